// SEPA_42434276884589
// MI455X (gfx1250) — compile-verified
//
#include <hip/hip_runtime.h>

// ---------------------------------------------------------------------------
// Graph auto-encoder forward for MI455X (gfx1250), all GEMMs on the CDNA5
// WMMA f32 path (V_WMMA_F32_16X16X4_F32), sparse aggregation via hardware
// global_atomic_add_f32. The pipeline is bound by the 256MB adj_recon store
// (~11us at 23.3 TB/s HBM), so f32 WMMA gives full accuracy with zero cost.
// ---------------------------------------------------------------------------

typedef __attribute__((ext_vector_type(2))) float v2f;
typedef __attribute__((ext_vector_type(8))) float v8f;

#define N_NODES 8192
#define N_EDGES 262144
#define D_FEAT  512
#define NHID    32
#define LATENT  16

// D = A(16x4 f32) * B(4x16 f32) + C(16x16 f32)
__device__ __forceinline__ v8f wmma4(v2f a, v2f b, v8f c) {
  return __builtin_amdgcn_wmma_f32_16x16x4_f32(
      /*neg_a=*/false, a, /*neg_b=*/false, b,
      /*c_mod=*/(short)0, c, /*reuse_a=*/false, /*reuse_b=*/false);
}

// ---------------------------------------------------------------------------
__global__ void k_zero(float* __restrict__ p, int n) {
  int i = blockIdx.x * blockDim.x + threadIdx.x;
  int stride = gridDim.x * blockDim.x;
  for (; i < n; i += stride) p[i] = 0.0f;
}

// xw1 = x @ Wg1            [8192,512]@[512,32]  (bias after aggregation)
// e1  = relu(x @ We1 + be1)
// grid 64 blocks x 256 thr = 512 waves, one 16-row tile per wave.
__global__ void __launch_bounds__(256) k_enc1(
    const float* __restrict__ x, const float* __restrict__ Wg1,
    const float* __restrict__ We1, const float* __restrict__ be1,
    float* __restrict__ xw1, float* __restrict__ e1) {
  const int lane = threadIdx.x & 31;
  const int tile = blockIdx.x * 8 + (threadIdx.x >> 5);  // 0..511
  const int row0 = tile * 16;
  const int lo = lane & 15, hi = lane >> 4;
  const float* Arow = x + (row0 + lo) * D_FEAT;
  v8f ag0 = {}, ag1 = {}, ae0 = {}, ae1 = {};
  for (int k0 = 0; k0 < D_FEAT; k0 += 4) {
    const int ka = k0 + 2 * hi;
    v2f a;  a.x  = Arow[ka];                 a.y  = Arow[ka + 1];
    v2f b0; b0.x = Wg1[ka*NHID + lo];        b0.y = Wg1[(ka+1)*NHID + lo];
    v2f b1; b1.x = Wg1[ka*NHID + 16 + lo];   b1.y = Wg1[(ka+1)*NHID + 16 + lo];
    v2f c0; c0.x = We1[ka*NHID + lo];        c0.y = We1[(ka+1)*NHID + lo];
    v2f c1; c1.x = We1[ka*NHID + 16 + lo];   c1.y = We1[(ka+1)*NHID + 16 + lo];
    ag0 = wmma4(a, b0, ag0);
    ag1 = wmma4(a, b1, ag1);
    ae0 = wmma4(a, c0, ae0);
    ae1 = wmma4(a, c1, ae1);
  }
  const float bl = be1[lo], bh = be1[16 + lo];
#pragma unroll
  for (int r = 0; r < 8; ++r) {
    const int m = row0 + r + 8 * hi;
    xw1[m*NHID + lo]      = ag0[r];
    xw1[m*NHID + 16 + lo] = ag1[r];
    float v0 = ae0[r] + bl, v1 = ae1[r] + bh;
    e1[m*NHID + lo]      = v0 > 0.0f ? v0 : 0.0f;
    e1[m*NHID + 16 + lo] = v1 > 0.0f ? v1 : 0.0f;
  }
}

// agg[dst] += val[src] * w   (W = 2^logw features; one wave covers 32/W edges
// with contiguous feature lanes -> coalesced loads + contiguous f32 atomics)
__global__ void k_scatter(const float* __restrict__ val,
                          const int* __restrict__ src,
                          const int* __restrict__ dst,
                          const float* __restrict__ ew,
                          float* __restrict__ agg, int logw) {
  const int tid = blockIdx.x * blockDim.x + threadIdx.x;
  const int e = tid >> logw;
  const int f = tid & ((1 << logw) - 1);
  const int W = 1 << logw;
  const float v = val[src[e] * W + f] * ew[e];
  __hip_atomic_fetch_add(&agg[dst[e] * W + f], v,
                         __ATOMIC_RELAXED, __HIP_MEMORY_SCOPE_AGENT);
}

__global__ void k_bias_relu(const float* __restrict__ in,
                            const float* __restrict__ b,
                            float* __restrict__ out, int mask) {
  const int i = blockIdx.x * blockDim.x + threadIdx.x;
  const float v = in[i] + b[i & mask];
  out[i] = v > 0.0f ? v : 0.0f;
}

// hw2 = h @ Wg2 (bias after aggregation);  zx = relu(e1 @ We2 + be2)
__global__ void __launch_bounds__(256) k_enc2(
    const float* __restrict__ h, const float* __restrict__ Wg2,
    const float* __restrict__ e1, const float* __restrict__ We2,
    const float* __restrict__ be2,
    float* __restrict__ hw2, float* __restrict__ zx) {
  const int lane = threadIdx.x & 31;
  const int tile = blockIdx.x * 8 + (threadIdx.x >> 5);
  const int row0 = tile * 16;
  const int lo = lane & 15, hi = lane >> 4;
  const float* Ah = h  + (row0 + lo) * NHID;
  const float* Ae = e1 + (row0 + lo) * NHID;
  v8f acc_h = {}, acc_e = {};
#pragma unroll
  for (int k0 = 0; k0 < NHID; k0 += 4) {
    const int ka = k0 + 2 * hi;
    v2f ah; ah.x = Ah[ka];                ah.y = Ah[ka + 1];
    v2f ae; ae.x = Ae[ka];                ae.y = Ae[ka + 1];
    v2f bh; bh.x = Wg2[ka*LATENT + lo];   bh.y = Wg2[(ka+1)*LATENT + lo];
    v2f be; be.x = We2[ka*LATENT + lo];   be.y = We2[(ka+1)*LATENT + lo];
    acc_h = wmma4(ah, bh, acc_h);
    acc_e = wmma4(ae, be, acc_e);
  }
  const float b = be2[lo];
#pragma unroll
  for (int r = 0; r < 8; ++r) {
    const int m = row0 + r + 8 * hi;
    hw2[m*LATENT + lo] = acc_h[r];
    const float v = acc_e[r] + b;
    zx[m*LATENT + lo] = v > 0.0f ? v : 0.0f;
  }
}

// za = relu(agg2 + bg2);  z = [za | zx]  written straight into d_out z slice
__global__ void k_zasm(const float* __restrict__ agg2,
                       const float* __restrict__ bg2,
                       const float* __restrict__ zx,
                       float* __restrict__ zout) {
  const int i = blockIdx.x * blockDim.x + threadIdx.x;  // [0, N*16)
  const int n = i >> 4, f = i & 15;
  const float v = agg2[i] + bg2[f];
  zout[n * 32 + f]      = v > 0.0f ? v : 0.0f;
  zout[n * 32 + 16 + f] = zx[i];
}

// adj = sigmoid(z @ z^T)  [8192,8192], K=32. grid(512,16) x 256thr:
// wave = one 16-row A tile cached in regs, 4 column tiles each.
__global__ void __launch_bounds__(256) k_adj(const float* __restrict__ z,
                                             float* __restrict__ adj) {
  const int lane = threadIdx.x & 31;
  const int rowTile = blockIdx.x;                              // 0..511
  const int cgrp = blockIdx.y * 8 + (threadIdx.x >> 5);        // 0..127
  const int lo = lane & 15, hi = lane >> 4;
  const float* Arow = z + (rowTile * 16 + lo) * 32;
  v2f a[8];
#pragma unroll
  for (int j = 0; j < 8; ++j) {
    const int ka = 4 * j + 2 * hi;
    a[j].x = Arow[ka]; a[j].y = Arow[ka + 1];
  }
  for (int t = 0; t < 4; ++t) {
    const int col0 = (cgrp * 4 + t) * 16;
    const float* Brow = z + (col0 + lo) * 32;   // B[k][n] = z[col0+n][k]
    v8f acc = {};
#pragma unroll
    for (int j = 0; j < 8; ++j) {
      const int ka = 4 * j + 2 * hi;
      v2f b; b.x = Brow[ka]; b.y = Brow[ka + 1];
      acc = wmma4(a[j], b, acc);
    }
#pragma unroll
    for (int r = 0; r < 8; ++r) {
      const float s = 1.0f / (1.0f + __expf(-acc[r]));
      adj[(long long)(rowTile * 16 + r + 8 * hi) * N_NODES + col0 + lo] = s;
    }
  }
}

// d1 = relu(zx @ Wd1 + bd1)   [8192,16]@[16,32]
__global__ void __launch_bounds__(256) k_dec1(
    const float* __restrict__ zx, const float* __restrict__ Wd1,
    const float* __restrict__ bd1, float* __restrict__ d1) {
  const int lane = threadIdx.x & 31;
  const int tile = blockIdx.x * 8 + (threadIdx.x >> 5);
  const int row0 = tile * 16;
  const int lo = lane & 15, hi = lane >> 4;
  const float* Arow = zx + (row0 + lo) * LATENT;
  v8f a0 = {}, a1 = {};
#pragma unroll
  for (int k0 = 0; k0 < LATENT; k0 += 4) {
    const int ka = k0 + 2 * hi;
    v2f a;  a.x  = Arow[ka];                a.y  = Arow[ka + 1];
    v2f b0; b0.x = Wd1[ka*NHID + lo];       b0.y = Wd1[(ka+1)*NHID + lo];
    v2f b1; b1.x = Wd1[ka*NHID + 16 + lo];  b1.y = Wd1[(ka+1)*NHID + 16 + lo];
    a0 = wmma4(a, b0, a0);
    a1 = wmma4(a, b1, a1);
  }
  const float bl = bd1[lo], bh = bd1[16 + lo];
#pragma unroll
  for (int r = 0; r < 8; ++r) {
    const int m = row0 + r + 8 * hi;
    float v0 = a0[r] + bl, v1 = a1[r] + bh;
    d1[m*NHID + lo]      = v0 > 0.0f ? v0 : 0.0f;
    d1[m*NHID + 16 + lo] = v1 > 0.0f ? v1 : 0.0f;
  }
}

// feat = relu(d1 @ Wd2 + bd2)  [8192,32]@[32,512]. grid(64,32)x256:
// rowTile = blockIdx.x*8+wave, colTile = blockIdx.y.
__global__ void __launch_bounds__(256) k_dec2(
    const float* __restrict__ d1, const float* __restrict__ Wd2,
    const float* __restrict__ bd2, float* __restrict__ feat) {
  const int lane = threadIdx.x & 31;
  const int rowTile = blockIdx.x * 8 + (threadIdx.x >> 5);  // 0..511
  const int colTile = blockIdx.y;                           // 0..31
  const int lo = lane & 15, hi = lane >> 4;
  const int row0 = rowTile * 16, col0 = colTile * 16;
  const float* Arow = d1 + (row0 + lo) * NHID;
  v8f acc = {};
#pragma unroll
  for (int k0 = 0; k0 < NHID; k0 += 4) {
    const int ka = k0 + 2 * hi;
    v2f a; a.x = Arow[ka];                      a.y = Arow[ka + 1];
    v2f b; b.x = Wd2[ka*D_FEAT + col0 + lo];    b.y = Wd2[(ka+1)*D_FEAT + col0 + lo];
    acc = wmma4(a, b, acc);
  }
  const float bias = bd2[col0 + lo];
#pragma unroll
  for (int r = 0; r < 8; ++r) {
    const float v = acc[r] + bias;
    feat[(row0 + r + 8 * hi) * D_FEAT + col0 + lo] = v > 0.0f ? v : 0.0f;
  }
}

// ---------------------------------------------------------------------------
extern "C" void kernel_launch(void* const* d_in, const int* in_sizes, int n_in,
                              void* d_out, int out_size, void* d_ws, size_t ws_size,
                              hipStream_t stream) {
  (void)in_sizes; (void)n_in; (void)out_size; (void)ws_size;

  const float* features = (const float*)d_in[0];
  const int*   esrc     = (const int*)d_in[1];
  const int*   edst     = (const int*)d_in[2];
  const float* ew       = (const float*)d_in[3];
  const float* Wg1      = (const float*)d_in[4];
  const float* bg1      = (const float*)d_in[5];
  const float* Wg2      = (const float*)d_in[6];
  const float* bg2      = (const float*)d_in[7];
  const float* We1      = (const float*)d_in[8];
  const float* be1      = (const float*)d_in[9];
  const float* We2      = (const float*)d_in[10];
  const float* be2      = (const float*)d_in[11];
  const float* Wd1      = (const float*)d_in[12];
  const float* bd1      = (const float*)d_in[13];
  const float* Wd2      = (const float*)d_in[14];
  const float* bd2      = (const float*)d_in[15];

  // workspace layout (floats), ~6.8 MB total
  float* ws   = (float*)d_ws;
  float* xw1  = ws;                 // [N,32]
  float* e1   = ws + 262144;        // [N,32]
  float* agg1 = ws + 524288;        // [N,32]  (zeroed; contiguous with agg2)
  float* agg2 = ws + 786432;        // [N,16]  (zeroed)
  float* h    = ws + 917504;        // [N,32]
  float* hw2  = ws + 1179648;       // [N,16]
  float* zx   = ws + 1310720;       // [N,16]
  float* d1b  = ws + 1441792;       // [N,32]

  float* adj  = (float*)d_out;                 // [8192,8192]
  float* feat = (float*)d_out + 67108864;      // [8192,512]
  float* zout = (float*)d_out + 71303168;      // [8192,32]

  k_zero     <<<512, 256, 0, stream>>>(agg1, 262144 + 131072);
  k_enc1     <<<64, 256, 0, stream>>>(features, Wg1, We1, be1, xw1, e1);
  k_scatter  <<<32768, 256, 0, stream>>>(xw1, esrc, edst, ew, agg1, 5);
  k_bias_relu<<<1024, 256, 0, stream>>>(agg1, bg1, h, 31);
  k_enc2     <<<64, 256, 0, stream>>>(h, Wg2, e1, We2, be2, hw2, zx);
  k_scatter  <<<16384, 256, 0, stream>>>(hw2, esrc, edst, ew, agg2, 4);
  k_zasm     <<<512, 256, 0, stream>>>(agg2, bg2, zx, zout);
  k_adj      <<<dim3(512, 16, 1), 256, 0, stream>>>(zout, adj);
  k_dec1     <<<64, 256, 0, stream>>>(zx, Wd1, bd1, d1b);
  k_dec2     <<<dim3(64, 32, 1), 256, 0, stream>>>(d1b, Wd2, bd2, feat);
}